// CRF_13563506721233
// MI455X (gfx1250) — compile-verified
//
#include <hip/hip_runtime.h>
#include <hip/hip_bf16.h>
#include <stdint.h>

#define S_LEN   2048
#define BATCH   256
#define C_N     64
#define NW      4            // waves per workgroup (prev-dimension split)
#define PREV_W  (C_N / NW)   // 16 prev states per wave
#define RING_N  16           // ring slots (issue depth is 8, slots 16 to avoid reuse races)
#define AHEAD   8            // async prefetch depth (steps)

// CDNA5 split wait for async global->LDS DMA (ASYNCcnt). Per-wave counter.
#define WAIT_ASYNCCNT(n) asm volatile("s_wait_asynccnt %0" :: "i"(n) : "memory")

// One step's emission row (256B): each of 32 lanes moves 8B global->LDS
// on the asynchronous DMA path, tracked with ASYNCcnt.
__device__ __forceinline__ void async_emit_load_b64(uint32_t lds_byte_off,
                                                    uint32_t voff_bytes,
                                                    uint64_t sbase) {
  asm volatile("global_load_async_to_lds_b64 %0, %1, %2"
               :: "v"(lds_byte_off), "v"(voff_bytes), "s"(sbase)
               : "memory");
}

__device__ __forceinline__ float rdlane(float v, int lane) {
  return __int_as_float(__builtin_amdgcn_readlane(__float_as_int(v), lane));
}

__global__ __launch_bounds__(NW * 32)
void viterbi_kernel(const float* __restrict__ em,
                    const float* __restrict__ mask,
                    const float* __restrict__ T,
                    float* __restrict__ out,
                    uint8_t* __restrict__ bp)
{
  const int b = blockIdx.x;
  const int l = threadIdx.x & 31;
  const int w = __builtin_amdgcn_readfirstlane(threadIdx.x) >> 5;  // uniform SGPR

  __shared__ float    ring[RING_N * C_N];      // emission staging ring (4KB)
  __shared__ float    pb[2][NW][C_N];          // partial best values  (2KB, dbl-buf)
  __shared__ uint16_t pi[2][NW][32];           // partial argmax pairs (512B)
  __shared__ float4   tile4[(64 * C_N) / 16];  // backtrace tile (4KB)

  // ---------------- sequence length = sum(mask[b,:]) (exact in f32) -------
  const float4* m4 = (const float4*)(mask + (size_t)b * S_LEN);
  float part = 0.f;
#pragma unroll
  for (int k = 0; k < (S_LEN / 4) / 32; ++k) {
    float4 mv = m4[k * 32 + l];
    part += (mv.x + mv.y) + (mv.z + mv.w);
  }
#pragma unroll
  for (int off = 16; off; off >>= 1) part += __shfl_xor(part, off, 32);
  const int len = (int)(part + 0.5f);          // in [S/2, S], same in all waves

  // ------- transition slice for this wave: prevs [16w, 16w+16) ------------
  // tt[pl] = { T[16w+pl][2l], T[16w+pl][2l+1] }  -> 32 VGPRs
  float2 tt[PREV_W];
  const float2* T2 = (const float2*)T;
#pragma unroll
  for (int pl = 0; pl < PREV_W; ++pl) tt[pl] = T2[(PREV_W * w + pl) * (C_N / 2) + l];

  // ---------------- alpha0 = emissions[:,0] + T[BOS=1,:] (all waves) ------
  const float2* em2 = (const float2*)(em + (size_t)b * S_LEN * C_N);
  float2 e0   = em2[l];
  float2 tbos = T2[1 * (C_N / 2) + l];         // T[BOS][2l], T[BOS][2l+1]
  float a0 = e0.x + tbos.x;                    // tag 2l   (mirrored in every wave)
  float a1 = e0.y + tbos.y;                    // tag 2l+1

  // ---------------- warm up the async emission ring (wave 0 only) ---------
  const uint64_t emBase   = (uint64_t)(uintptr_t)(em + (size_t)b * S_LEN * C_N);
  const uint32_t ringBase = (uint32_t)(uintptr_t)ring;   // LDS byte offset
  if (w == 0) {
#pragma unroll
    for (int t = 1; t <= AHEAD; ++t) {
      async_emit_load_b64(ringBase + (uint32_t)(((t & (RING_N - 1)) * C_N + 2 * l) * 4),
                          (uint32_t)((t * C_N + 2 * l) * 4), emBase);
    }
  }

  uint8_t* bpRow = bp + (size_t)b * S_LEN * C_N;
  volatile float* ring_v = ring;               // async engine writes, not the compiler

  // ---------------- forward recursion (max-plus, 4-way prev split) --------
  for (int t = 1; t < len; ++t) {
    const int slot = t & (RING_N - 1);
    const int dbuf = t & 1;

    // partial max/argmax over this wave's 16 prev states, for tags 2l, 2l+1
    float best0 = -INFINITY, best1 = -INFINITY;
    int   i0 = 0, i1 = 0;
#pragma unroll
    for (int pl = 0; pl < PREV_W; ++pl) {
      const int p = PREV_W * w + pl;           // global prev index (SGPR)
      float src = (pl & 1) ? a1 : a0;          // alpha[p] lives in lane p>>1
      float ap  = rdlane(src, 8 * w + (pl >> 1));
      float s0 = ap + tt[pl].x;
      float s1 = ap + tt[pl].y;
      if (s0 > best0) { best0 = s0; i0 = p; }  // strict > = first occurrence
      if (s1 > best1) { best1 = s1; i1 = p; }
    }
    float2 pv; pv.x = best0; pv.y = best1;
    *(float2*)&pb[dbuf][w][2 * l] = pv;
    pi[dbuf][w][l] = (uint16_t)(i0 | (i1 << 8));

    if (w == 0) WAIT_ASYNCCNT(AHEAD - 1);      // step t's emissions are in LDS
    __syncthreads();                           // partials + ring visible to all

    // refill: slot (t+8)&15 was consumed at t-8 (8 barriers ago) -> race-free
    if (w == 0) {
      int tp = t + AHEAD; if (tp > S_LEN - 1) tp = S_LEN - 1;  // always 1 issue
      async_emit_load_b64(ringBase + (uint32_t)((((t + AHEAD) & (RING_N - 1)) * C_N + 2 * l) * 4),
                          (uint32_t)((tp * C_N + 2 * l) * 4), emBase);
    }

    // every wave redundantly merges the 4 partials -> identical alpha mirror
    float2   q0 = *(const float2*)&pb[dbuf][0][2 * l];
    uint32_t j0 = pi[dbuf][0][l];
    float m0 = q0.x, m1 = q0.y;
    int  mi0 = (int)(j0 & 63), mi1 = (int)(j0 >> 8);
#pragma unroll
    for (int wp = 1; wp < NW; ++wp) {          // ascending wave = ascending prev range
      float2   q = *(const float2*)&pb[dbuf][wp][2 * l];
      uint32_t j = pi[dbuf][wp][l];
      if (q.x > m0) { m0 = q.x; mi0 = (int)(j & 63); }
      if (q.y > m1) { m1 = q.y; mi1 = (int)(j >> 8); }
    }

    float ex = ring_v[slot * C_N + 2 * l];
    float ey = ring_v[slot * C_N + 2 * l + 1];
    a0 = m0 + ex;
    a1 = m1 + ey;

    if (w == 0)                                // backpointers into position t
      *(uint16_t*)(bpRow + (size_t)t * C_N + 2 * l) = (uint16_t)(mi0 | (mi1 << 8));
  }

  if (w != 0) return;                          // no barriers past this point
  WAIT_ASYNCCNT(0);                            // drain stray prefetches

  // ---------------- terminal: alpha += T[:,EOS=2]; argmax -----------------
  float f0 = a0 + T[(2 * l) * C_N + 2];
  float f1 = a1 + T[(2 * l + 1) * C_N + 2];
  float v;  int bi;
  if (f0 >= f1) { v = f0; bi = 2 * l; } else { v = f1; bi = 2 * l + 1; }
#pragma unroll
  for (int off = 16; off; off >>= 1) {
    float ov = __shfl_xor(v, off, 32);
    int   oi = __shfl_xor(bi, off, 32);
    if (ov > v || (ov == v && oi < bi)) { v = ov; bi = oi; }
  }
  // bi is now uniform best_last

  // ---------------- outputs: tail PAD + last tag --------------------------
  float* orow = out + (size_t)b * S_LEN;
  for (int t = len + l; t < S_LEN; t += 32) orow[t] = 0.0f;   // PAD
  if (l == 0) orow[len - 1] = (float)bi;

  // ---------------- backtrace via 4KB LDS tiles ---------------------------
  asm volatile("s_wait_storecnt 0" ::: "memory");  // bp stores visible to our loads
  uint8_t* tileB = (uint8_t*)tile4;
  int cur = bi;
  int hi  = len - 1;
  while (hi >= 1) {
    int loadLo = hi - 63; if (loadLo < 0) loadLo = 0;
    int lo     = hi - 63; if (lo < 1)     lo = 1;
    const float4* src = (const float4*)(bpRow + (size_t)loadLo * C_N);
#pragma unroll
    for (int c = 0; c < 8; ++c) tile4[c * 32 + l] = src[c * 32 + l];
    asm volatile("s_wait_dscnt 0" ::: "memory");
    for (int pos = hi; pos >= lo; --pos) {     // uniform broadcast chain in LDS
      cur = tileB[(pos - loadLo) * C_N + cur];
      if (l == 0) orow[pos - 1] = (float)cur;
    }
    hi = lo - 1;
  }
}

extern "C" void kernel_launch(void* const* d_in, const int* in_sizes, int n_in,
                              void* d_out, int out_size, void* d_ws, size_t ws_size,
                              hipStream_t stream) {
  const float* em    = (const float*)d_in[0];   // [B,S,C] f32
  const float* mask  = (const float*)d_in[1];   // [B,S]   f32
  const float* trans = (const float*)d_in[2];   // [C,C]   f32
  float*   out = (float*)d_out;                 // [B,S] tag ids as f32
  uint8_t* bpw = (uint8_t*)d_ws;                // needs B*S*C = 32 MiB scratch
  (void)in_sizes; (void)n_in; (void)out_size; (void)ws_size;
  viterbi_kernel<<<dim3(BATCH), dim3(NW * 32), 0, stream>>>(em, mask, trans, out, bpw);
}